// GraphSage_22127671509058
// MI455X (gfx1250) — compile-verified
//
#include <hip/hip_runtime.h>

// ---------------------------------------------------------------------------
// Fused 2-layer GraphSAGE forward for MI455X (gfx1250, wave32).
//
// Roofline: ~700MB of random feature gathers (~30us @ 23.3TB/s, less with the
// 192MB L2 holding most of the 205MB feature table) vs ~38.7 GFLOP of exact
// fp32 matmul via V_WMMA_F32_16X16X4_F32 -> roughly balanced, so we keep fp32
// precision instead of down-converting. h1 (272x256 per block) lives entirely
// in the 320KB/WGP LDS -> saves the 285MB HBM round-trip of a 2-kernel split.
// ---------------------------------------------------------------------------

typedef __attribute__((ext_vector_type(2))) float v2f;
typedef __attribute__((ext_vector_type(4))) float v4f;
typedef __attribute__((ext_vector_type(8))) float v8f;

#define FDIM   256          // feature dim
#define KDIM   512          // concat dim (2*F)
#define FAN1   16
#define FAN2   4
#define NPB    16           // batch nodes per block
#define ROWS   272          // 17*16 layer-1 rows per block (1 self + 16 neigh per node)
#define MT     17           // M tiles of 16
#define KC     64           // K chunk staged in LDS
#define THREADS 512         // 16 wave32s
#define B_TOTAL 8192

// LDS layout (floats), phases separated by barriers:
//  phase1 staging : Xs  @ 0          [272][68]   (73,984 B)
//                   Ws  @ 272*68     [256][68]   (69,632 B)   end 143,616 B
//  h1 (after GEMM): H1  @ 0          [272][260]  (282,880 B)  overwrites staging
//  layer-2 input  : X2  @ 272*260    [16][516]   ( 33,024 B)  end 315,904 B
//  phase2 staging : Ws2 @ 0          [256][68]   (h1 dead by then)
#define XS_OFF   0
#define WS_OFF   (272 * 68)
#define H1_OFF   0
#define H1_STRIDE 260
#define X2_OFF   (272 * 260)
#define X2_STRIDE 516
#define WS2_OFF  0
#define SMEM_FLOATS (272 * 260 + 16 * 516)   // 78,976 floats = 315,904 B

__global__ __launch_bounds__(THREADS)
void sage_fused_kernel(const float* __restrict__ feats,
                       const float* __restrict__ W1,
                       const float* __restrict__ W2,
                       const int*   __restrict__ bn,     // batch_nodes [B]
                       const int*   __restrict__ ng1,    // neigh1 [B][16]
                       const int*   __restrict__ ng2,    // neigh2 [M1][4]
                       float*       __restrict__ out)    // [B][256]
{
    extern __shared__ float smem[];
    float* Xs  = smem + XS_OFF;
    float* Ws  = smem + WS_OFF;
    float* H1  = smem + H1_OFF;
    float* X2  = smem + X2_OFF;
    float* Ws2 = smem + WS2_OFF;

    const int tid  = threadIdx.x;
    const int wave = tid >> 5;        // 0..15 -> owns N-tile `wave`
    const int lane = tid & 31;
    const int lm   = lane & 15;       // lane % 16
    const int lh   = lane >> 4;       // 0/1: K-half for A/B frags, M-half for C/D
    const int b0   = blockIdx.x * NPB;

    // ---------------- Layer 1: h1 = relu([self | mean4(neigh2)] @ W1^T) -----
    v8f acc[MT];
    #pragma unroll
    for (int i = 0; i < MT; ++i)
        #pragma unroll
        for (int r = 0; r < 8; ++r) acc[i][r] = 0.0f;

    for (int kc = 0; kc < KDIM; kc += KC) {
        // Stage X chunk: 272 rows x 64 cols (as float4), gather + mean fused.
        for (int t = tid; t < ROWS * 16; t += THREADS) {
            const int row = t >> 4;
            const int col = (t & 15) * 4;
            const int k   = kc + col;
            const int gi  = (row < NPB) ? (b0 + row)
                                        : (B_TOTAL + b0 * FAN1 + (row - NPB));
            v4f val;
            if (k < FDIM) {
                int node;
                if (row < NPB) {
                    node = bn[b0 + row];
                } else {
                    const int r = row - NPB;
                    node = ng1[(b0 + (r >> 4)) * FAN1 + (r & 15)];
                }
                val = *(const v4f*)(feats + (size_t)node * FDIM + k);
            } else {
                const int kk = k - FDIM;
                const int* nb = ng2 + (size_t)gi * FAN2;
                v4f a0 = *(const v4f*)(feats + (size_t)nb[0] * FDIM + kk);
                v4f a1 = *(const v4f*)(feats + (size_t)nb[1] * FDIM + kk);
                v4f a2 = *(const v4f*)(feats + (size_t)nb[2] * FDIM + kk);
                v4f a3 = *(const v4f*)(feats + (size_t)nb[3] * FDIM + kk);
                val = (a0 + a1 + a2 + a3) * 0.25f;
            }
            *(v4f*)(Xs + row * 68 + col) = val;
        }
        // Stage W1 chunk: 256 rows (n) x 64 cols (k), coalesced.
        for (int t = tid; t < 256 * 16; t += THREADS) {
            const int row = t >> 4;
            const int col = (t & 15) * 4;
            *(v4f*)(Ws + row * 68 + col) =
                *(const v4f*)(W1 + (size_t)row * KDIM + kc + col);
        }
        __syncthreads();

        // 16 K-steps of 4; wave owns N-tile `wave`, all 17 M-tiles.
        #pragma unroll 4
        for (int ks = 0; ks < KC; ks += 4) {
            const v2f bfrag =
                *(const v2f*)(Ws + (wave * 16 + lm) * 68 + ks + 2 * lh);
            #pragma unroll
            for (int mt = 0; mt < MT; ++mt) {
                const v2f afrag =
                    *(const v2f*)(Xs + (mt * 16 + lm) * 68 + ks + 2 * lh);
                acc[mt] = __builtin_amdgcn_wmma_f32_16x16x4_f32(
                    false, afrag, false, bfrag, (short)0, acc[mt],
                    false, false);
            }
        }
        __syncthreads();
    }

    // ReLU + write h1 to LDS (overwrites staging region; barriers around it).
    #pragma unroll
    for (int mt = 0; mt < MT; ++mt) {
        #pragma unroll
        for (int r = 0; r < 8; ++r) {
            const int row = mt * 16 + r + 8 * lh;
            const float v = acc[mt][r];
            H1[row * H1_STRIDE + wave * 16 + lm] = v > 0.0f ? v : 0.0f;
        }
    }
    __syncthreads();

    // ---------------- Build X2 = [h1_self | mean16(h1_neigh)] --------------
    for (int t = tid; t < NPB * (KDIM / 4); t += THREADS) {
        const int i = t >> 7;          // node within block
        const int k = (t & 127) * 4;
        v4f val;
        if (k < FDIM) {
            val = *(const v4f*)(H1 + i * H1_STRIDE + k);
        } else {
            const int kk = k - FDIM;
            v4f s;
            #pragma unroll
            for (int r = 0; r < 4; ++r) s[r] = 0.0f;
            #pragma unroll
            for (int j = 0; j < FAN1; ++j)
                s += *(const v4f*)(H1 + (NPB + i * FAN1 + j) * H1_STRIDE + kk);
            val = s * (1.0f / 16.0f);
        }
        *(v4f*)(X2 + i * X2_STRIDE + k) = val;
    }
    __syncthreads();   // h1 dead after this point -> Ws2 may overwrite it

    // ---------------- Layer 2: out = relu(X2 @ W2^T) -----------------------
    v8f acc2;
    #pragma unroll
    for (int r = 0; r < 8; ++r) acc2[r] = 0.0f;

    for (int kc = 0; kc < KDIM; kc += KC) {
        for (int t = tid; t < 256 * 16; t += THREADS) {
            const int row = t >> 4;
            const int col = (t & 15) * 4;
            *(v4f*)(Ws2 + row * 68 + col) =
                *(const v4f*)(W2 + (size_t)row * KDIM + kc + col);
        }
        __syncthreads();

        #pragma unroll 4
        for (int ks = 0; ks < KC; ks += 4) {
            const v2f afrag =
                *(const v2f*)(X2 + lm * X2_STRIDE + kc + ks + 2 * lh);
            const v2f bfrag =
                *(const v2f*)(Ws2 + (wave * 16 + lm) * 68 + ks + 2 * lh);
            acc2 = __builtin_amdgcn_wmma_f32_16x16x4_f32(
                false, afrag, false, bfrag, (short)0, acc2, false, false);
        }
        __syncthreads();
    }

    // Store: D VGPR r -> out row b0 + r + 8*lh, col wave*16 + lm.
    #pragma unroll
    for (int r = 0; r < 8; ++r) {
        const float v = acc2[r];
        out[(size_t)(b0 + r + 8 * lh) * FDIM + wave * 16 + lm] =
            v > 0.0f ? v : 0.0f;
    }
}

extern "C" void kernel_launch(void* const* d_in, const int* in_sizes, int n_in,
                              void* d_out, int out_size, void* d_ws, size_t ws_size,
                              hipStream_t stream) {
    (void)in_sizes; (void)n_in; (void)out_size; (void)d_ws; (void)ws_size;
    const float* feats = (const float*)d_in[0];
    const float* W1    = (const float*)d_in[1];
    const float* W2    = (const float*)d_in[2];
    const int*   bn    = (const int*)d_in[3];
    const int*   ng1   = (const int*)d_in[4];
    const int*   ng2   = (const int*)d_in[5];
    float*       out   = (float*)d_out;

    const size_t smem_bytes = (size_t)SMEM_FLOATS * sizeof(float); // 315,904 B
    // Allow >64KB dynamic LDS (CDNA5 WGP has 320KB).
    hipFuncSetAttribute((const void*)sage_fused_kernel,
                        hipFuncAttributeMaxDynamicSharedMemorySize,
                        (int)smem_bytes);

    const int blocks = B_TOTAL / NPB;   // 512 blocks of 512 threads (16 waves)
    sage_fused_kernel<<<blocks, THREADS, smem_bytes, stream>>>(
        feats, W1, W2, bn, ng1, ng2, out);
}